// OctreeDWConvBn_4286377361900
// MI455X (gfx1250) — compile-verified
//
#include <hip/hip_runtime.h>
#include <cstdint>
#include <cstddef>

#define C_CH      192
#define C2        96          // float2 channel pairs == conv block size (3 waves)
#define K_TAPS    27
#define NODE_TILE 64
#define BN_BLOCK  192
#define BN_EPS    1e-5f

typedef uint32_t u32x4 __attribute__((ext_vector_type(4)));
typedef uint32_t u32x8 __attribute__((ext_vector_type(8)));

__device__ __forceinline__ uint32_t lds_offset_of(const void* p) {
  return (uint32_t)(uintptr_t)p;   // flat addr low 32 bits == LDS byte offset
}

// ---------------------------------------------------------------------------
// Kernel A: zero the 2*C_CH stats accumulators in workspace
// ---------------------------------------------------------------------------
__global__ void __launch_bounds__(2 * C_CH)
zero_stats_kernel(float* __restrict__ stats) {
  const int t = threadIdx.x;
  if (t < 2 * C_CH) stats[t] = 0.0f;
}

// ---------------------------------------------------------------------------
// Kernel B: depthwise gathered conv + per-block partial BN sums.
//   96 threads = one float2 channel-pair per thread (covers all 192 channels);
//   every thread walks all NODE_TILE rows -> row index is block-uniform, so
//   gathered row bases go through readfirstlane -> scalar (saddr) addressing.
//   The 64x27 neighbor-index tile is staged into LDS by ONE Tensor Data Mover
//   DMA (tensor_load_to_lds, TENSORcnt), issued by wave 0 only.
// ---------------------------------------------------------------------------
__global__ void __launch_bounds__(C2)
octree_dwconv_kernel(const float* __restrict__ data,
                     const int*   __restrict__ neigh,
                     const float* __restrict__ weight,
                     float*       __restrict__ out,
                     float*       __restrict__ stats,
                     int n_nodes) {
  __shared__ int s_idx[NODE_TILE * K_TAPS];   // 64*27*4 = 6912 B

  const int c2    = threadIdx.x;              // 0..95 -> channels 2*c2, 2*c2+1
  const int node0 = blockIdx.x * NODE_TILE;
  int rows = n_nodes - node0;
  if (rows > NODE_TILE) rows = NODE_TILE;

  // ---- TDM: DMA the contiguous index tile into LDS (wave 0 only) ---------
  if (threadIdx.x < 32) {
    const uint64_t ga    = (uint64_t)(uintptr_t)(neigh + (size_t)node0 * K_TAPS);
    const uint32_t lds0  = lds_offset_of(&s_idx[0]);
    const uint32_t total = (uint32_t)(rows * K_TAPS);   // <= 1728 elements

    u32x4 g0;
    g0.x = 1u;                                          // count=1 (valid D#)
    g0.y = lds0;                                        // lds_addr (bytes)
    g0.z = (uint32_t)ga;                                // global_addr[31:0]
    g0.w = ((uint32_t)(ga >> 32) & 0x01FFFFFFu)         // global_addr[56:32]
           | 0x80000000u;                               // type = 2 ("image")

    u32x8 g1;
    g1.s0 = 2u << 16;                                   // data_size = 4 bytes
    g1.s1 = (total & 0xFFFFu) << 16;                    // tensor_dim0[15:0]
    g1.s2 = ((total >> 16) & 0xFFFFu) | (1u << 16);     // tensor_dim0[31:16], tensor_dim1 = 1
    g1.s3 = (total & 0xFFFFu) << 16;                    // tile_dim0 = total (1-D tile)
    g1.s4 = 0u;                                         // tile_dim1 = tile_dim2 = 0 (unused)
    g1.s5 = total;                                      // tensor_dim0_stride[31:0]
    g1.s6 = 0u;
    g1.s7 = 0u;

    asm volatile("tensor_load_to_lds %0, %1" : : "s"(g0), "s"(g1) : "memory");
    __builtin_amdgcn_s_wait_tensorcnt(0);               // drain this wave's TENSORcnt
  }
  __syncthreads();                                      // LDS tile visible to all waves

  // ---- Per-pair weight columns into VGPRs (coalesced b64, once) ----------
  float2 w[K_TAPS];
#pragma unroll
  for (int k = 0; k < K_TAPS; ++k) {
    w[k] = *reinterpret_cast<const float2*>(weight + k * C_CH + 2 * c2);
  }

  // ---- Gathered depthwise conv over the node tile ------------------------
  float2 lsum = make_float2(0.f, 0.f);
  float2 lsq  = make_float2(0.f, 0.f);

#pragma unroll 2
  for (int r = 0; r < rows; ++r) {
    float2 acc = make_float2(0.f, 0.f);
#pragma unroll
    for (int k = 0; k < K_TAPS; ++k) {
      // block-uniform neighbor index -> scalar row base (saddr addressing)
      const int nb = __builtin_amdgcn_readfirstlane(s_idx[r * K_TAPS + k]);
      const float2 v =
          *reinterpret_cast<const float2*>(data + (size_t)nb * C_CH + 2 * c2);
      acc.x = fmaf(v.x, w[k].x, acc.x);
      acc.y = fmaf(v.y, w[k].y, acc.y);
    }
    *reinterpret_cast<float2*>(out + (size_t)(node0 + r) * C_CH + 2 * c2) = acc;
    lsum.x += acc.x;
    lsum.y += acc.y;
    lsq.x = fmaf(acc.x, acc.x, lsq.x);
    lsq.y = fmaf(acc.y, acc.y, lsq.y);
  }

  // ---- Block partial sums -> global stats (native f32 atomics) -----------
  const int c = 2 * c2;
  unsafeAtomicAdd(&stats[c],            lsum.x);
  unsafeAtomicAdd(&stats[c + 1],        lsum.y);
  unsafeAtomicAdd(&stats[C_CH + c],     lsq.x);
  unsafeAtomicAdd(&stats[C_CH + c + 1], lsq.y);
}

// ---------------------------------------------------------------------------
// Kernel C: BatchNorm apply (training-mode batch statistics), in place
// ---------------------------------------------------------------------------
__global__ void __launch_bounds__(BN_BLOCK)
bn_apply_kernel(float*       __restrict__ out,
                const float* __restrict__ stats,
                const float* __restrict__ gamma,
                const float* __restrict__ beta,
                int n_nodes, float inv_n) {
  const int t  = threadIdx.x;
  const int c2 = t % C2;
  const int g  = t / C2;
  const int c  = 2 * c2;

  const float2 sum = *reinterpret_cast<const float2*>(stats + c);
  const float2 sq  = *reinterpret_cast<const float2*>(stats + C_CH + c);
  const float2 gm  = *reinterpret_cast<const float2*>(gamma + c);
  const float2 bt  = *reinterpret_cast<const float2*>(beta + c);

  const float mean_x = sum.x * inv_n;
  const float mean_y = sum.y * inv_n;
  const float var_x  = fmaf(-mean_x, mean_x, sq.x * inv_n);
  const float var_y  = fmaf(-mean_y, mean_y, sq.y * inv_n);
  const float sc_x   = rsqrtf(var_x + BN_EPS) * gm.x;
  const float sc_y   = rsqrtf(var_y + BN_EPS) * gm.y;
  const float sh_x   = fmaf(-mean_x, sc_x, bt.x);
  const float sh_y   = fmaf(-mean_y, sc_y, bt.y);

  const int node0 = blockIdx.x * NODE_TILE;
  int rows = n_nodes - node0;
  if (rows > NODE_TILE) rows = NODE_TILE;

  for (int r = g; r < rows; r += 2) {
    float2* p =
        reinterpret_cast<float2*>(out + (size_t)(node0 + r) * C_CH + c);
    float2 v = *p;
    v.x = fmaf(v.x, sc_x, sh_x);
    v.y = fmaf(v.y, sc_y, sh_y);
    *p = v;
  }
}

// ---------------------------------------------------------------------------
// Host launcher
// ---------------------------------------------------------------------------
extern "C" void kernel_launch(void* const* d_in, const int* in_sizes, int n_in,
                              void* d_out, int out_size, void* d_ws, size_t ws_size,
                              hipStream_t stream) {
  const float* data   = (const float*)d_in[0];  // [N, C]
  const int*   neigh  = (const int*)  d_in[1];  // [N, K]
  const float* weight = (const float*)d_in[2];  // [K, C]
  const float* gamma  = (const float*)d_in[3];  // [C]
  const float* beta   = (const float*)d_in[4];  // [C]
  float* out   = (float*)d_out;                 // [N, C]
  float* stats = (float*)d_ws;                  // [2*C] partial sums

  const int n_nodes = in_sizes[1] / K_TAPS;
  const int nblocks = (n_nodes + NODE_TILE - 1) / NODE_TILE;

  zero_stats_kernel<<<1, 2 * C_CH, 0, stream>>>(stats);
  octree_dwconv_kernel<<<nblocks, C2, 0, stream>>>(data, neigh, weight,
                                                   out, stats, n_nodes);
  bn_apply_kernel<<<nblocks, BN_BLOCK, 0, stream>>>(out, stats, gamma, beta,
                                                    n_nodes, 1.0f / (float)n_nodes);
}